// Decoder_91233695302179
// MI455X (gfx1250) — compile-verified
//
#include <hip/hip_runtime.h>
#include <math.h>

#define B_TOT 2048
#define T_LEN 128
#define S_LEN 256
#define VOCAB 29
#define EMB   32
#define HID   64
#define KX    160      // EMB + HID(context) + HID(h)
#define XS    164      // X row stride (floats), 16B-aligned rows, conflict-skewed
#define CS    132      // combined row stride
#define HS    68       // h/c row stride
#define GS    257      // scores/gates row stride

typedef __attribute__((ext_vector_type(2))) float v2f;
typedef __attribute__((ext_vector_type(4))) float v4f;
typedef __attribute__((ext_vector_type(8))) float v8f;

// Fast activations built only from TRANS ops (v_exp_f32 / v_rcp_f32) — keeps the
// recurrent critical path free of IEEE divide fixup sequences.
__device__ __forceinline__ float fast_rcp(float x)   { return __builtin_amdgcn_rcpf(x); }
__device__ __forceinline__ float sigmoidf_(float x)  { return fast_rcp(1.0f + __expf(-x)); }
__device__ __forceinline__ float tanhf_(float x) {
    // tanh(x) = 1 - 2/(exp(2x)+1); exp->inf gives rcp->0 => +1, exp->0 => -1.
    return 1.0f - 2.0f * fast_rcp(__expf(2.0f * x) + 1.0f);
}

__global__ __launch_bounds__(256) void decoder_attn_lstm_kernel(
    const int*   __restrict__ y,
    const float* __restrict__ h0,
    const float* __restrict__ c0,
    const float* __restrict__ enc,
    const float* __restrict__ emb_table,
    const float* __restrict__ W_ih,
    const float* __restrict__ W_hh,
    const float* __restrict__ b_ih,
    const float* __restrict__ b_hh,
    const float* __restrict__ fc_W,
    const float* __restrict__ fc_b,
    float*       __restrict__ out)
{
    __shared__ __align__(16) float sh_h[16 * HS];      // hidden state, persists across t
    __shared__ __align__(16) float sh_c[16 * HS];      // cell state
    __shared__ __align__(16) float sh_sg[16 * GS];     // scores (attn) then gates (aliased)
    __shared__ __align__(16) float sh_X[16 * XS];      // A matrix for gates WMMA: [emb|ctx|h]
    __shared__ __align__(16) float sh_comb[16 * CS];   // A matrix for logits WMMA: [h_new|ctx]
    __shared__               float sh_red[16 * 16];
    __shared__               float sh_stat[16];

    const int tid   = threadIdx.x;
    const int lane  = tid & 31;
    const int wave  = tid >> 5;
    const int lhalf = lane >> 4;      // 0: lanes 0-15, 1: lanes 16-31
    const int l15   = lane & 15;
    const int bg    = blockIdx.x * 16;

    // init h, c from h0/c0
    for (int idx = tid; idx < 16 * HID; idx += 256) {
        int b = idx >> 6, j = idx & 63;
        sh_h[b * HS + j] = h0[(bg + b) * HID + j];
        sh_c[b * HS + j] = c0[(bg + b) * HID + j];
    }
    // warm weight caches (global_prefetch_b8); weights are reused every step
    if (tid < 96)  __builtin_prefetch(W_ih + tid * 256, 0, 3);
    if (tid < 64)  __builtin_prefetch(W_hh + tid * 256, 0, 3);
    if (tid < 15)  __builtin_prefetch(fc_W + tid * 256, 0, 3);
    __syncthreads();

    const int rb = tid >> 4;          // batch row 0..15 (16-thread teams)
    const int rj = tid & 15;

    for (int t = 0; t < T_LEN; ++t) {
        // ---- 1) attention scores: sg[b][s] = enc[b][s][:] . h[b][:] (batched matvec, VALU)
        {
            const float* encb = enc + (size_t)(bg + rb) * S_LEN * HID;
            for (int s = rj; s < S_LEN; s += 16) {
                const v4f* er = (const v4f*)(encb + s * HID);
                const v4f* hr = (const v4f*)(sh_h + rb * HS);
                float acc = 0.f;
                #pragma unroll
                for (int q = 0; q < 16; ++q) {
                    v4f e = er[q], hh = hr[q];
                    acc += e.x * hh.x + e.y * hh.y + e.z * hh.z + e.w * hh.w;
                }
                sh_sg[rb * GS + s] = acc;
            }
        }
        __syncthreads();

        // ---- 2) softmax over s per batch row (16-thread team per row)
        {
            float m = -3.4e38f;
            for (int s = rj; s < S_LEN; s += 16) m = fmaxf(m, sh_sg[rb * GS + s]);
            sh_red[rb * 16 + rj] = m;
            __syncthreads();
            if (rj == 0) {
                float mm = sh_red[rb * 16];
                #pragma unroll
                for (int q = 1; q < 16; ++q) mm = fmaxf(mm, sh_red[rb * 16 + q]);
                sh_stat[rb] = mm;
            }
            __syncthreads();
            float rowm = sh_stat[rb];
            float sum = 0.f;
            for (int s = rj; s < S_LEN; s += 16) {
                float e = __expf(sh_sg[rb * GS + s] - rowm);
                sh_sg[rb * GS + s] = e;
                sum += e;
            }
            __syncthreads();   // all exp writes done before stat overwrite readers
            sh_red[rb * 16 + rj] = sum;
            __syncthreads();
            if (rj == 0) {
                float ss = 0.f;
                #pragma unroll
                for (int q = 0; q < 16; ++q) ss += sh_red[rb * 16 + q];
                sh_stat[rb] = fast_rcp(ss);
            }
            __syncthreads();
        }

        // ---- 3) context: ctx[b][h] = sum_s attn * enc (4 h-values per thread, coalesced)
        {
            float inv = sh_stat[rb];
            int h4 = rj * 4;
            const float* encb = enc + (size_t)(bg + rb) * S_LEN * HID + h4;
            v4f acc = {0.f, 0.f, 0.f, 0.f};
            for (int s = 0; s < S_LEN; ++s) {
                float a = sh_sg[rb * GS + s];
                v4f e = *(const v4f*)(encb + (size_t)s * HID);
                acc.x += a * e.x; acc.y += a * e.y; acc.z += a * e.z; acc.w += a * e.w;
            }
            acc.x *= inv; acc.y *= inv; acc.z *= inv; acc.w *= inv;
            *(v4f*)(sh_X    + rb * XS + EMB + h4) = acc;   // X ctx region
            *(v4f*)(sh_comb + rb * CS + HID + h4) = acc;   // combined ctx region
        }

        // ---- 4) fill X emb region (embedding gather) and X h region (h_{t-1})
        for (int idx = tid; idx < 16 * EMB; idx += 256) {
            int b = idx >> 5, k = idx & 31;
            int tok = y[(bg + b) * T_LEN + t];
            sh_X[b * XS + k] = emb_table[tok * EMB + k];
        }
        for (int idx = tid; idx < 16 * HID; idx += 256) {
            int b = idx >> 6, j = idx & 63;
            sh_X[b * XS + EMB + HID + j] = sh_h[b * HS + j];
        }
        __syncthreads();

        // ---- 5) gates GEMM via WMMA: [16 x 160] @ [160 x 256] -> sh_sg (aliased)
        {
            const int n0 = wave * 32;                 // each wave: 2 N-tiles of 16
            v8f acc0 = {}; v8f acc1 = {};
            const float* Xrow = sh_X + l15 * XS;
            const int na = n0 + l15;
            const int nb = na + 16;
            #pragma unroll 4
            for (int k0 = 0; k0 < KX; k0 += 4) {
                int ka = k0 + (lhalf << 1);           // A/B K-pair per lane half
                v2f a;
                a.x = Xrow[ka];
                a.y = Xrow[ka + 1];
                v2f bva, bvb;
                if (ka < EMB + HID) {                  // W_ih covers k in [0,96)
                    bva = *(const v2f*)(W_ih + na * (EMB + HID) + ka);
                    bvb = *(const v2f*)(W_ih + nb * (EMB + HID) + ka);
                } else {                               // W_hh covers k in [96,160)
                    int kh = ka - (EMB + HID);
                    bva = *(const v2f*)(W_hh + na * HID + kh);
                    bvb = *(const v2f*)(W_hh + nb * HID + kh);
                }
                acc0 = __builtin_amdgcn_wmma_f32_16x16x4_f32(
                    false, a, false, bva, (short)0, acc0, false, false);
                acc1 = __builtin_amdgcn_wmma_f32_16x16x4_f32(
                    false, a, false, bvb, (short)0, acc1, false, false);
            }
            int mrow = lhalf ? 8 : 0;                 // C/D layout: VGPR r -> M=r (+8 for hi half)
            #pragma unroll
            for (int r = 0; r < 8; ++r) {
                sh_sg[(mrow + r) * GS + na]      = acc0[r];
                sh_sg[(mrow + r) * GS + na + 16] = acc1[r];
            }
        }
        __syncthreads();

        // ---- 6) LSTM elementwise (gate order i, f, g, o) + write h_new into combined
        for (int idx = tid; idx < 16 * HID; idx += 256) {
            int b = idx >> 6, j = idx & 63;
            float gi = sh_sg[b * GS +        j] + b_ih[j]       + b_hh[j];
            float gf = sh_sg[b * GS +  64 + j] + b_ih[64 + j]   + b_hh[64 + j];
            float gg = sh_sg[b * GS + 128 + j] + b_ih[128 + j]  + b_hh[128 + j];
            float go = sh_sg[b * GS + 192 + j] + b_ih[192 + j]  + b_hh[192 + j];
            float cn = sigmoidf_(gf) * sh_c[b * HS + j] + sigmoidf_(gi) * tanhf_(gg);
            float hn = sigmoidf_(go) * tanhf_(cn);
            sh_c[b * HS + j] = cn;
            sh_h[b * HS + j] = hn;
            sh_comb[b * CS + j] = hn;
        }
        __syncthreads();

        // ---- 7) logits GEMM via WMMA: [16 x 128] @ [128 x 32(pad of 29)] -> out
        if (wave < 2) {
            const int n = wave * 16 + l15;
            const bool nv = (n < VOCAB);
            v8f acc = {};
            const float* Crow = sh_comb + l15 * CS;
            #pragma unroll 4
            for (int k0 = 0; k0 < 2 * HID; k0 += 4) {
                int ka = k0 + (lhalf << 1);
                v2f a;
                a.x = Crow[ka];
                a.y = Crow[ka + 1];
                v2f bv;
                bv.x = nv ? fc_W[n * (2 * HID) + ka]     : 0.f;
                bv.y = nv ? fc_W[n * (2 * HID) + ka + 1] : 0.f;
                acc = __builtin_amdgcn_wmma_f32_16x16x4_f32(
                    false, a, false, bv, (short)0, acc, false, false);
            }
            if (nv) {
                float bb = fc_b[n];
                int mrow = lhalf ? 8 : 0;
                #pragma unroll
                for (int r = 0; r < 8; ++r) {
                    int bl = mrow + r;
                    out[((size_t)(bg + bl) * T_LEN + t) * VOCAB + n] = acc[r] + bb;
                }
            }
        }
        __syncthreads();   // protect sh_comb / sh_sg reuse next iteration
    }
}

extern "C" void kernel_launch(void* const* d_in, const int* in_sizes, int n_in,
                              void* d_out, int out_size, void* d_ws, size_t ws_size,
                              hipStream_t stream) {
    (void)in_sizes; (void)n_in; (void)out_size; (void)d_ws; (void)ws_size;
    const int*   y         = (const int*)  d_in[0];
    const float* h0        = (const float*)d_in[1];
    const float* c0        = (const float*)d_in[2];
    const float* enc       = (const float*)d_in[3];
    const float* emb_table = (const float*)d_in[4];
    const float* W_ih      = (const float*)d_in[5];
    const float* W_hh      = (const float*)d_in[6];
    const float* b_ih      = (const float*)d_in[7];
    const float* b_hh      = (const float*)d_in[8];
    const float* fc_W      = (const float*)d_in[9];
    const float* fc_b      = (const float*)d_in[10];
    float* out = (float*)d_out;

    dim3 grid(B_TOT / 16);   // 128 blocks, each owns 16 batch rows for all T steps
    dim3 block(256);          // 8 wave32s
    decoder_attn_lstm_kernel<<<grid, block, 0, stream>>>(
        y, h0, c0, enc, emb_table, W_ih, W_hh, b_ih, b_hh, fc_W, fc_b, out);
}